// incepLayer_20667382628951
// MI455X (gfx1250) — compile-verified
//
#include <hip/hip_runtime.h>

// Problem constants (match reference setup_inputs)
#define NN 40000      // nodes
#define NE 640000     // edges
// D = 128 features = 32 lanes x float4 ; out row = 512 floats

#define WPB 8         // waves per block (spmm)
#define TPB 256

// ---- CDNA5 async global->LDS path (guarded; fallback = shfl broadcast) ----
#if defined(__has_builtin)
#if __has_builtin(__builtin_amdgcn_global_load_async_to_lds_b64)
#define HAVE_ASYNC_COPY 1
#endif
#if __has_builtin(__builtin_amdgcn_s_wait_asynccnt)
#define HAVE_WAIT_ASYNC 1
#endif
#endif

#define AS1 __attribute__((address_space(1)))
#define AS3 __attribute__((address_space(3)))

typedef int   v2i __attribute__((ext_vector_type(2)));
typedef float v4f __attribute__((ext_vector_type(4)));

#if defined(HAVE_ASYNC_COPY)
__device__ __forceinline__ void async_copy_edge_b64(const int2* g, int2* l) {
  // signature: (v2i AS1* gsrc, v2i AS3* ldst, imm offset, imm cpol)
  __builtin_amdgcn_global_load_async_to_lds_b64((AS1 v2i*)g, (AS3 v2i*)l, 0, 0);
}
__device__ __forceinline__ void wait_async0() {
#if defined(HAVE_WAIT_ASYNC)
  __builtin_amdgcn_s_wait_asynccnt(0);
#else
  asm volatile("s_wait_asynccnt 0" ::: "memory");
#endif
  asm volatile("" ::: "memory"); // compiler ordering barrier
}
#endif

// ------------------------- small helpers -------------------------
__device__ __forceinline__ float4 f4z() { return make_float4(0.f, 0.f, 0.f, 0.f); }
__device__ __forceinline__ float4 f4fma(float s, float4 a, float4 b) { // s*a + b
  return make_float4(fmaf(s, a.x, b.x), fmaf(s, a.y, b.y),
                     fmaf(s, a.z, b.z), fmaf(s, a.w, b.w));
}
__device__ __forceinline__ float4 f4scale(float s, float4 a) {
  return make_float4(s * a.x, s * a.y, s * a.z, s * a.w);
}
// non-temporal float4 store: output slices are write-once streams; keep them
// from evicting the L2-resident gather working set (h/m1/m2/CSR).
// __builtin_nontemporal_store needs a native clang vector type -> bit-cast.
__device__ __forceinline__ void nt_store_f4(float4* p, float4 v) {
  v4f t;
  t.x = v.x; t.y = v.y; t.z = v.z; t.w = v.w;
  __builtin_nontemporal_store(t, (v4f*)p);
}

// ------------------------- CSR build -------------------------
__global__ void k_zero_i32(int* __restrict__ p, int n) {
  int i = blockIdx.x * blockDim.x + threadIdx.x;
  if (i < n) p[i] = 0;
}

__global__ void k_hist(const int* __restrict__ dst, int* __restrict__ cnt, int e) {
  int i = blockIdx.x * blockDim.x + threadIdx.x;
  if (i < e) atomicAdd(&cnt[dst[i]], 1);
}

// single-block in-place exclusive scan over n ints (n <= 1024*40 here)
__global__ __launch_bounds__(1024) void k_scan(int* __restrict__ data, int n) {
  __shared__ int s[1024];
  const int T = 1024;
  int t = threadIdx.x;
  int chunk = (n + T - 1) / T;
  int lo = t * chunk;
  int hi = lo + chunk; if (hi > n) hi = n;
  int sum = 0;
  for (int i = lo; i < hi; ++i) sum += data[i];
  s[t] = sum;
  __syncthreads();
  for (int off = 1; off < T; off <<= 1) {   // Hillis-Steele inclusive scan
    int v = (t >= off) ? s[t - off] : 0;
    __syncthreads();
    s[t] += v;
    __syncthreads();
  }
  int run = (t == 0) ? 0 : s[t - 1];
  for (int i = lo; i < hi; ++i) { int c = data[i]; data[i] = run; run += c; }
}

// pack edges into CSR order: csr[p] = { src, bits(d[src]*d[dst]) }
__global__ void k_scatter(const int* __restrict__ src, const int* __restrict__ dst,
                          const float* __restrict__ dvec,
                          const int* __restrict__ rowp, int* __restrict__ fill,
                          int2* __restrict__ csr, int e) {
  int i = blockIdx.x * blockDim.x + threadIdx.x;
  if (i >= e) return;
  int v = dst[i], u = src[i];
  int p = rowp[v] + atomicAdd(&fill[v], 1);
  csr[p] = make_int2(u, __float_as_int(dvec[u] * dvec[v]));
}

// ------------------------- fused SpMM + combine -------------------------
// One wave32 per node; lane holds 4 features (float4). acc = sum_e w_e * X[src_e].
// mode 1: out[:,128:256] = a0*acc + (1-a0)*h ; also copies h into out[:,0:128]
// mode 2: out[:,256:384] = poly(acc=m2, m1=X, h)  (coeffs from a1,a2)
// mode 3: out[:,384:512] = poly(acc=m3, m2=X, m1=M1, h)  (coeffs from a3,a4,a5)
__global__ __launch_bounds__(TPB) void k_spmm(
    const int* __restrict__ rowp, const int2* __restrict__ csr,
    const float4* __restrict__ X, float4* __restrict__ Mout,
    float4* __restrict__ out, const float* __restrict__ alphas,
    const float4* __restrict__ H, const float4* __restrict__ M1, int mode) {
#if defined(HAVE_ASYNC_COPY)
  __shared__ int2 ebuf[WPB][32];
#endif
  const int lane = threadIdx.x & 31;
  const int wib  = threadIdx.x >> 5;
  const int node = blockIdx.x * WPB + wib;
  if (node >= NN) return;

  const int beg = rowp[node];
  const int end = rowp[node + 1];
  float4 acc = f4z();

  for (int base = beg; base < end; base += 32) {
    int cnt = end - base; if (cnt > 32) cnt = 32;
#if defined(HAVE_ASYNC_COPY)
    // stage this wave's edge records into LDS with the async tensor/LDS path
    if (lane < cnt) async_copy_edge_b64(&csr[base + lane], &ebuf[wib][lane]);
    wait_async0();
#pragma unroll 4
    for (int j = 0; j < cnt; ++j) {
      int2 r = ebuf[wib][j];                 // uniform DS broadcast
      float w = __int_as_float(r.y);
      float4 x = X[r.x * 32 + lane];         // coalesced 128b row gather
      acc = f4fma(w, x, acc);
    }
#else
    int2 rec = make_int2(0, 0);
    if (lane < cnt) rec = csr[base + lane];  // coalesced per-lane edge load
#pragma unroll 4
    for (int j = 0; j < cnt; ++j) {
      int   u = __shfl(rec.x, j, 32);
      float w = __int_as_float(__shfl(rec.y, j, 32));
      float4 x = X[u * 32 + lane];
      acc = f4fma(w, x, acc);
    }
#endif
  }

  const int rix = node * 32 + lane;
  if (Mout) Mout[rix] = acc;              // regular store: consumed by next SpMM

  const float4 h = H[rix];
  const int obase = node * 128;              // out row = 128 float4 (512 floats)

  if (mode == 1) {
    float a0 = alphas[0];
    nt_store_f4(&out[obase + lane], h);                      // slice 0 = h
    nt_store_f4(&out[obase + 32 + lane],
                f4fma(a0, acc, f4scale(1.f - a0, h)));       // slice 1 = P1
  } else if (mode == 2) {
    float a1 = alphas[1], a2 = alphas[2];
    float c2 = a2 * a1;
    float c1 = a2 * (1.f - a1) + (1.f - a2) * a1;
    float c0 = (1.f - a2) * (1.f - a1);
    float4 m1v = X[rix];
    float4 o = f4fma(c2, acc, f4fma(c1, m1v, f4scale(c0, h)));
    nt_store_f4(&out[obase + 64 + lane], o);                 // slice 2 = P2
  } else {
    float a3 = alphas[3], a4 = alphas[4], a5 = alphas[5];
    float u1 = a3, u0 = 1.f - a3;
    float v2 = a4 * u1;
    float v1 = a4 * u0 + (1.f - a4) * u1;
    float v0 = (1.f - a4) * u0;
    float c3 = a5 * v2;
    float c2 = a5 * v1 + (1.f - a5) * v2;
    float c1 = a5 * v0 + (1.f - a5) * v1;
    float c0 = (1.f - a5) * v0;
    float4 m2v = X[rix];
    float4 m1v = M1[rix];
    float4 o = f4fma(c3, acc,
               f4fma(c2, m2v, f4fma(c1, m1v, f4scale(c0, h))));
    nt_store_f4(&out[obase + 96 + lane], o);                 // slice 3 = P3
  }
}

// ------------------------- host launcher -------------------------
extern "C" void kernel_launch(void* const* d_in, const int* in_sizes, int n_in,
                              void* d_out, int out_size, void* d_ws, size_t ws_size,
                              hipStream_t stream) {
  (void)in_sizes; (void)n_in; (void)out_size; (void)ws_size;
  const float* h      = (const float*)d_in[0];   // [NN,128]
  const float* dvec   = (const float*)d_in[1];   // [NN]
  const float* alphas = (const float*)d_in[2];   // [6]
  const int*   src    = (const int*)d_in[3];     // [NE]
  const int*   dst    = (const int*)d_in[4];     // [NE]
  float* out = (float*)d_out;                    // [NN,512]

  // workspace layout
  char* ws = (char*)d_ws;
  constexpr size_t O_ROWP = 0;                                   // (NN+1) ints
  constexpr size_t O_FILL = O_ROWP + (size_t)(NN + 1) * 4;       // NN ints
  constexpr size_t O_CSR  = ((O_FILL + (size_t)NN * 4) + 15) & ~(size_t)15; // NE int2
  constexpr size_t O_M1   = ((O_CSR + (size_t)NE * 8) + 15) & ~(size_t)15;  // NN*128 f32
  constexpr size_t O_M2   = O_M1 + (size_t)NN * 128 * 4;
  int*  rowp = (int*)(ws + O_ROWP);
  int*  fill = (int*)(ws + O_FILL);
  int2* csr  = (int2*)(ws + O_CSR);
  float4* m1 = (float4*)(ws + O_M1);
  float4* m2 = (float4*)(ws + O_M2);

  const float4* h4 = (const float4*)h;
  float4* out4 = (float4*)out;

  // 1) build CSR (counts -> scan -> scatter), zeroed fresh every call
  int nz = 2 * NN + 1;                                  // rowp + fill contiguous
  k_zero_i32<<<(nz + 255) / 256, 256, 0, stream>>>(rowp, nz);
  k_hist<<<(NE + 255) / 256, 256, 0, stream>>>(dst, rowp, NE);
  k_scan<<<1, 1024, 0, stream>>>(rowp, NN + 1);
  k_scatter<<<(NE + 255) / 256, 256, 0, stream>>>(src, dst, dvec, rowp, fill, csr, NE);

  // 2) three Krylov SpMMs with fused output combines
  const int blocks = (NN + WPB - 1) / WPB;              // 5000
  k_spmm<<<blocks, TPB, 0, stream>>>(rowp, csr, h4, m1, out4, alphas, h4, nullptr, 1);
  k_spmm<<<blocks, TPB, 0, stream>>>(rowp, csr, m1, m2, out4, alphas, h4, nullptr, 2);
  k_spmm<<<blocks, TPB, 0, stream>>>(rowp, csr, m2, nullptr, out4, alphas, h4, m1, 3);
}